// ConvSoftArgmax3d_2353642078936
// MI455X (gfx1250) — compile-verified
//
#include <hip/hip_runtime.h>
#include <cstdint>

// Problem constants (from reference: x = (8,16,4,256,256) f32, k=3,s=1,p=1)
#define BC   128            // b*c
#define Dd   4
#define Hh   256
#define Ww   256
#define HW   (Hh * Ww)      // 65536
#define DHW  (Dd * HW)      // 262144
#define TX   32             // output tile width
#define TY   16             // output tile height
#define HX   (TX + 2)       // halo tile width  = 34
#define HY   (TY + 2)       // halo tile height = 18
#define NSEG 4              // pass-1 segments per channel

static constexpr size_t CO_TOTAL = (size_t)BC * 3 * DHW;  // coords element count

typedef __attribute__((ext_vector_type(4))) unsigned int u32x4;
typedef __attribute__((ext_vector_type(4))) int          i32x4;
typedef __attribute__((ext_vector_type(8))) int          i32x8;

// ---------------------------------------------------------------------------
// TDM: DMA a 3D tile (clipW x clipH x 4 planes) of f32 from global into LDS.
// Descriptor layout per CDNA5 ISA ch.8 (D# groups 0..3), data_size = 4B,
// tensor_dim0_stride = W, tensor_dim1_stride = H*W. Tile is pre-clamped to
// the image so no TDM OOB handling is needed.
// ---------------------------------------------------------------------------
__device__ inline void tdm_load_tile(const float* gaddr, unsigned lds_off,
                                     int clipW, int clipH) {
#if __has_builtin(__builtin_amdgcn_tensor_load_to_lds)
  unsigned long long ga = (unsigned long long)(uintptr_t)gaddr;
  u32x4 g0;
  g0[0] = 1u;                       // [1:0] count=1 (valid user descriptor)
  g0[1] = lds_off;                  // [63:32] lds_addr (bytes)
  g0[2] = (unsigned)ga;             // [95:64] global_addr[31:0]
  g0[3] = (unsigned)((ga >> 32) & 0x1FFFFFFull) | (2u << 30); // addr[56:32], type=2
  i32x8 g1;
  g1[0] = 2 << 16;                  // data_size = 2 (4 bytes)
  g1[1] = clipW << 16;              // tensor_dim0[15:0] at bits[63:48]
  g1[2] = clipH << 16;              // tensor_dim0 hi=0 | tensor_dim1[15:0]
  g1[3] = clipW << 16;              // tensor_dim1 hi=0 | tile_dim0
  g1[4] = clipH | (Dd << 16);       // tile_dim1 | tile_dim2
  g1[5] = Ww;                       // tensor_dim0_stride[31:0] = 256
  g1[6] = (HW & 0xFFFF) << 16;      // stride0 hi=0 | stride1[15:0] (=0)
  g1[7] = HW >> 16;                 // stride1[47:16] (65536 -> 1)
  i32x4 g2;
  g2[0] = Dd;                       // tensor_dim2
  g2[1] = 1;                        // tensor_dim3 (unused, tile_dim3=0)
  g2[2] = 0;                        // tensor_dim2_stride lo (unused)
  g2[3] = 0;                        // stride hi | tile_dim3 = 0
  i32x4 g3 = {0, 0, 0, 0};
  i32x8 g4 = {0, 0, 0, 0, 0, 0, 0, 0};   // extra operand on clang-23 lane
  __builtin_amdgcn_tensor_load_to_lds(g0, g1, g2, g3, g4, 0);
#else
  (void)gaddr; (void)lds_off; (void)clipW; (void)clipH;
#endif
}

// ---------------------------------------------------------------------------
// Pass 2: soft-argmax 3x3x3 stencil. One block = one (bc, 16x32) tile,
// all 4 output z-slices (so the 4 input planes are staged exactly once).
// Defined FIRST in the file so the disasm snippet shows the TDM issue path.
// ---------------------------------------------------------------------------
__global__ __launch_bounds__(256) void softargmax3d(const float* __restrict__ x,
                                                    const float* __restrict__ part,
                                                    float* __restrict__ out) {
  const int tid = threadIdx.x;
  const int tx0 = blockIdx.x * TX;
  const int ty0 = blockIdx.y * TY;
  const int bc  = blockIdx.z;

  __shared__ float  xraw[Dd * HY * HX];         // raw x tile (clamped, packed)
  __shared__ float2 ep[(Dd + 2) * HY * HX];     // {e, e*x}, zero-padded halo vol

  // Clamp halo tile to the image (left/top halos shrink at the borders)
  int ys = ty0 - 1; if (ys < 0) ys = 0;
  int ye = ty0 + TY + 1; if (ye > Hh) ye = Hh;
  int xs = tx0 - 1; if (xs < 0) xs = 0;
  int xe = tx0 + TX + 1; if (xe > Ww) xe = Ww;
  const int clipH = ye - ys;
  const int clipW = xe - xs;

  const float* gbase = x + (size_t)bc * DHW + (size_t)ys * Ww + (size_t)xs;

#if __has_builtin(__builtin_amdgcn_tensor_load_to_lds)
  if (tid == 0)
    tdm_load_tile(gbase, (unsigned)(uintptr_t)&xraw[0], clipW, clipH);
  __builtin_amdgcn_s_wait_tensorcnt(0);   // unconditional: no-op for waves w/ cnt==0
#else
  // Fallback staging (also what the host pass sees): same packed layout.
  {
    const int tot = Dd * clipH * clipW;
    for (int i = tid; i < tot; i += 256) {
      int pz = i / (clipH * clipW);
      int r  = i - pz * clipH * clipW;
      int py = r / clipW;
      int px = r - py * clipW;
      xraw[i] = gbase[(size_t)pz * HW + (size_t)py * Ww + (size_t)px];
    }
  }
#endif
  __syncthreads();

  const float* pm = part + bc * NSEG;
  const float mx = fmaxf(fmaxf(pm[0], pm[1]), fmaxf(pm[2], pm[3]));

  // Build zero-padded {e, p} halo volume (pad applies to x_exp, matching ref).
  const int CELLS = (Dd + 2) * HY * HX;
  for (int i = tid; i < CELLS; i += 256) {
    int lz = i / (HY * HX);
    int r  = i - lz * (HY * HX);
    int ly = r / HX;
    int lx = r - ly * HX;
    int gz = lz - 1;
    int gy = ty0 - 1 + ly;
    int gx = tx0 - 1 + lx;
    float e = 0.0f, p = 0.0f;
    if ((unsigned)gz < (unsigned)Dd && (unsigned)gy < (unsigned)Hh &&
        (unsigned)gx < (unsigned)Ww) {
      float xv = xraw[(gz * clipH + (gy - ys)) * clipW + (gx - xs)];
      e = __expf(xv - mx);   // TEMP = 1.0
      p = e * xv;
    }
    ep[i] = make_float2(e, p);   // one ds_store_b64
  }
  __syncthreads();

  // 2048 output points per block -> 8 per thread. 27-point b64 gather from LDS.
  for (int k = 0; k < (Dd * TY * TX) / 256; ++k) {
    int pid = tid + k * 256;
    int lx = pid & (TX - 1);
    int ly = (pid >> 5) & (TY - 1);
    int zo = pid >> 9;
    float S = 0.f, Sx = 0.f, Sy = 0.f, Sz = 0.f, Sv = 0.f;
#pragma unroll
    for (int dz = 0; dz < 3; ++dz) {
#pragma unroll
      for (int dy = 0; dy < 3; ++dy) {
        const float2* er = &ep[((zo + dz) * HY + (ly + dy)) * HX + lx];
        float2 a = er[0], b = er[1], c = er[2];
        float srow = a.x + b.x + c.x;
        S  += srow;
        Sx += c.x - a.x;                   // dx weight in {-1,0,1}
        Sy += (float)(dy - 1) * srow;      // dy weight
        Sz += (float)(dz - 1) * srow;      // dz weight
        Sv += a.y + b.y + c.y;
      }
    }
    float inv = 1.0f / (S + 1e-8f);        // den = sumpool(e) + eps
    int gy = ty0 + ly, gx = tx0 + lx;
    size_t plane = (size_t)zo * HW + (size_t)gy * Ww + (size_t)gx;
    size_t cb = (size_t)bc * 3 * DHW;
    // coords channel order (z, x, y); grid_pooled == global index (center k)
    __builtin_nontemporal_store(Sz * inv + (float)zo, out + cb + plane);
    __builtin_nontemporal_store(Sx * inv + (float)gx, out + cb + (size_t)DHW + plane);
    __builtin_nontemporal_store(Sy * inv + (float)gy, out + cb + 2 * (size_t)DHW + plane);
    __builtin_nontemporal_store(Sv * inv, out + CO_TOTAL + (size_t)bc * DHW + plane);
  }
}

// ---------------------------------------------------------------------------
// Pass 1: per-(b,c) partial max over 1/4 of the channel volume.
// Regular-temporal loads intentionally warm L2 (x = 128MB < 192MB L2) so the
// pass-2 TDM reads mostly hit L2.
// ---------------------------------------------------------------------------
__global__ __launch_bounds__(256) void chmax_part(const float* __restrict__ x,
                                                  float* __restrict__ part) {
  const int seg = blockIdx.x & (NSEG - 1);
  const int bc  = blockIdx.x >> 2;
  const float4* p =
      (const float4*)(x + (size_t)bc * DHW + (size_t)seg * (DHW / NSEG));
  const int n4 = DHW / NSEG / 4;  // 16384 float4 per segment
  float m = -3.402823466e38f;
  for (int i = threadIdx.x; i < n4; i += 256) {
    float4 v = p[i];
    m = fmaxf(m, fmaxf(fmaxf(v.x, v.y), fmaxf(v.z, v.w)));
  }
  __shared__ float red[256];
  red[threadIdx.x] = m;
  __syncthreads();
  for (int s = 128; s > 0; s >>= 1) {
    if (threadIdx.x < s)
      red[threadIdx.x] = fmaxf(red[threadIdx.x], red[threadIdx.x + s]);
    __syncthreads();
  }
  if (threadIdx.x == 0) part[blockIdx.x] = red[0];
}

extern "C" void kernel_launch(void* const* d_in, const int* in_sizes, int n_in,
                              void* d_out, int out_size, void* d_ws, size_t ws_size,
                              hipStream_t stream) {
  (void)in_sizes; (void)n_in; (void)out_size; (void)ws_size;
  const float* x   = (const float*)d_in[0];
  float* out       = (float*)d_out;
  float* part      = (float*)d_ws;   // 512 floats of scratch, rewritten each call

  chmax_part<<<dim3(BC * NSEG), dim3(256), 0, stream>>>(x, part);
  softargmax3d<<<dim3(Ww / TX, Hh / TY, BC), dim3(256), 0, stream>>>(x, part, out);
}